// LlamaAttention_78812649882214
// MI455X (gfx1250) — compile-verified
//
#include <hip/hip_runtime.h>
#include <hip/hip_bf16.h>
#include <stddef.h>

// ---------------------------------------------------------------------------
// MI455X (gfx1250) Llama attention layer, fp32 end to end.
// All GEMMs + attention matmuls go through V_WMMA_F32_16X16X4_F32.
// N=1, L=2048, DIM=2048, H=32, HD=64.
//
// Round-2 changes (driven by round-1 asm: s_wait_loadcnt 0x0 before every
// wmma = zero overlap):
//   * GEMM: 32Mx64N wave tile (8 wmma / 6 loads per k-step, B-frag reuse)
//     + software pipeline: next k-step fragments loaded before current wmmas.
//   * Attention: all 16 K-fragments hoisted ahead of the S wmma chain;
//     V-fragments issued before the softmax VALU work (latency hidden by
//     v_exp_f32 + shuffles instead of stalling the P@V wmmas).
// ---------------------------------------------------------------------------

#define LSEQ 2048
#define DMODEL 2048
#define NHEADS 32
#define HDIM 64

typedef __attribute__((ext_vector_type(2))) float v2f;
typedef __attribute__((ext_vector_type(8))) float v8f;

// D = A(16x4) * B(4x16) + C  (fp32 WMMA, wave32)
__device__ __forceinline__ v8f wmma4(v2f a, v2f b, v8f c) {
  return __builtin_amdgcn_wmma_f32_16x16x4_f32(
      /*neg_a=*/false, a, /*neg_b=*/false, b,
      /*c_mod=*/(short)0, c, /*reuse_a=*/false, /*reuse_b=*/false);
}

// ---------------------------------------------------------------------------
// GEMM: C[M][N] = A[MxK] * B[NxK]^T   (row-major fp32; "x @ W.T" pattern)
// Wave tile 32Mx64N, workgroup = 8 waves arranged 2Mx4N -> 64Mx256N.
// A-frag (16x4): lane m(0-15) holds A[m][k0..k0+1]; lanes 16-31 hold k0+2..3.
// B-frag (4x16): lane n(0-15) holds B[k0..k0+1][n] = Bmat[n][k0..k0+1].
// Software-pipelined: loads for k-step i+1 issue before the wmmas of step i.
// ---------------------------------------------------------------------------
__global__ __launch_bounds__(256)
void gemm_nt_f32(const float* __restrict__ A, const float* __restrict__ B,
                 float* __restrict__ C, int M, int N, int K)
{
  const int lane = threadIdx.x & 31;
  const int wave = threadIdx.x >> 5;
  const int wm = wave & 1;        // 0..1
  const int wn = wave >> 1;       // 0..3
  const int m0 = blockIdx.x * 64 + wm * 32;
  const int n0 = blockIdx.y * 256 + wn * 64;
  const int half = lane >> 4;     // 0 or 1
  const int l16 = lane & 15;

  const float* pa0 = A + (size_t)(m0 + l16) * K + 2 * half;
  const float* pa1 = pa0 + (size_t)16 * K;
  const float* pb0 = B + (size_t)(n0 + l16) * K + 2 * half;
  const float* pb1 = pb0 + (size_t)16 * K;
  const float* pb2 = pb0 + (size_t)32 * K;
  const float* pb3 = pb0 + (size_t)48 * K;

  v8f acc00{}, acc01{}, acc02{}, acc03{};
  v8f acc10{}, acc11{}, acc12{}, acc13{};

  // prologue: fragments for k=0
  v2f a0 = *(const v2f*)(pa0);
  v2f a1 = *(const v2f*)(pa1);
  v2f b0 = *(const v2f*)(pb0);
  v2f b1 = *(const v2f*)(pb1);
  v2f b2 = *(const v2f*)(pb2);
  v2f b3 = *(const v2f*)(pb3);

  for (int k0 = 4; k0 < K; k0 += 4) {
    // issue next k-step loads FIRST (in flight behind current wmmas)
    v2f a0n = *(const v2f*)(pa0 + k0);
    v2f a1n = *(const v2f*)(pa1 + k0);
    v2f b0n = *(const v2f*)(pb0 + k0);
    v2f b1n = *(const v2f*)(pb1 + k0);
    v2f b2n = *(const v2f*)(pb2 + k0);
    v2f b3n = *(const v2f*)(pb3 + k0);

    acc00 = wmma4(a0, b0, acc00);
    acc01 = wmma4(a0, b1, acc01);
    acc02 = wmma4(a0, b2, acc02);
    acc03 = wmma4(a0, b3, acc03);
    acc10 = wmma4(a1, b0, acc10);
    acc11 = wmma4(a1, b1, acc11);
    acc12 = wmma4(a1, b2, acc12);
    acc13 = wmma4(a1, b3, acc13);

    a0 = a0n; a1 = a1n;
    b0 = b0n; b1 = b1n; b2 = b2n; b3 = b3n;
  }
  // epilogue: last k-step
  acc00 = wmma4(a0, b0, acc00);
  acc01 = wmma4(a0, b1, acc01);
  acc02 = wmma4(a0, b2, acc02);
  acc03 = wmma4(a0, b3, acc03);
  acc10 = wmma4(a1, b0, acc10);
  acc11 = wmma4(a1, b1, acc11);
  acc12 = wmma4(a1, b2, acc12);
  acc13 = wmma4(a1, b3, acc13);

  // C/D layout: vgpr r -> row r (lanes 0-15) / row r+8 (lanes 16-31), col=l16.
  float* c0 = C + (size_t)(m0 + 8 * half) * N + n0 + l16;
  float* c1 = C + (size_t)(m0 + 16 + 8 * half) * N + n0 + l16;
#pragma unroll
  for (int r = 0; r < 8; ++r) {
    c0[0]  = acc00[r];
    c0[16] = acc01[r];
    c0[32] = acc02[r];
    c0[48] = acc03[r];
    c1[0]  = acc10[r];
    c1[16] = acc11[r];
    c1[32] = acc12[r];
    c1[48] = acc13[r];
    c0 += N; c1 += N;
  }
}

// ---------------------------------------------------------------------------
// RoPE for Q (interleaved-pair variant of reference), pre-scaled by hd^-0.5.
// q_raw: (L, DMODEL) row-major ; out: (H, L, HDIM)
// rot[t] = (t<32) ? -q[2t+1] : q[2(t-32)]
// ---------------------------------------------------------------------------
__global__ __launch_bounds__(256)
void rope_q_kernel(const float* __restrict__ q_raw,
                   const float* __restrict__ cosb,   // (L, HDIM)
                   const float* __restrict__ sinb,   // (L, HDIM)
                   float* __restrict__ q_out)        // (H, L, HDIM)
{
  int idx = blockIdx.x * 256 + threadIdx.x;     // h*(L*64) + l*64 + t
  int t = idx & 63;
  int l = (idx >> 6) & (LSEQ - 1);
  int h = idx >> 17;
  const float* qr = q_raw + (size_t)l * DMODEL + h * HDIM;
  float x   = qr[t];
  float rot = (t < 32) ? -qr[2 * t + 1] : qr[2 * (t - 32)];
  float val = x * cosb[l * HDIM + t] + rot * sinb[l * HDIM + t];
  q_out[idx] = val * 0.125f;                    // 1/sqrt(64)
}

// ---------------------------------------------------------------------------
// RoPE for K (transposed layout) + cache blend. Writes final k output.
// k_raw: (L, DMODEL); cosT/sinT: (HDIM, L); k_cache/k_out: (H, HDIM, L);
// k_mask: (L)
// ---------------------------------------------------------------------------
__global__ __launch_bounds__(256)
void rope_k_merge_kernel(const float* __restrict__ k_raw,
                         const float* __restrict__ cosT,
                         const float* __restrict__ sinT,
                         const float* __restrict__ k_cache,
                         const float* __restrict__ k_mask,
                         float* __restrict__ k_out)
{
  int idx = blockIdx.x * 256 + threadIdx.x;     // h*(64*L) + t*L + l
  int l = idx & (LSEQ - 1);
  int t = (idx >> 11) & 63;
  int h = idx >> 17;
  const float* kr = k_raw + (size_t)l * DMODEL + h * HDIM;
  float x   = kr[t];
  float rot = (t < 32) ? -kr[2 * t + 1] : kr[2 * (t - 32)];
  float krot = x * cosT[t * LSEQ + l] + rot * sinT[t * LSEQ + l];
  float km = k_mask[l];
  k_out[idx] = k_cache[idx] * (1.0f - km) + krot * km;
}

// ---------------------------------------------------------------------------
// V cache blend. v_raw: (L, DMODEL); v_cache/v_out: (H, L, HDIM); v_mask: (L)
// ---------------------------------------------------------------------------
__global__ __launch_bounds__(256)
void v_merge_kernel(const float* __restrict__ v_raw,
                    const float* __restrict__ v_cache,
                    const float* __restrict__ v_mask,
                    float* __restrict__ v_out)
{
  int idx = blockIdx.x * 256 + threadIdx.x;     // h*(L*64) + l*64 + t
  int t = idx & 63;
  int l = (idx >> 6) & (LSEQ - 1);
  int h = idx >> 17;
  float vv = v_raw[(size_t)l * DMODEL + h * HDIM + t];
  float vm = v_mask[l];
  v_out[idx] = v_cache[idx] * (1.0f - vm) + vv * vm;
}

// ---------------------------------------------------------------------------
// Flash-style attention. One wave owns a 16-row Q block of one head and
// streams K/V in 16-column blocks with online softmax.
//   Q : (H, L, HDIM)   pre-scaled by hd^-0.5
//   Kt: (H, HDIM, L)   (lane-coalesced B-fragments for Q @ K)
//   V : (H, L, HDIM)
//   mask: (L, L)
//   Y : (L, DMODEL)    head h writes columns [h*64, h*64+64)
// ---------------------------------------------------------------------------
__global__ __launch_bounds__(256)
void attn_kernel(const float* __restrict__ Q,
                 const float* __restrict__ Kt,
                 const float* __restrict__ V,
                 const float* __restrict__ mask,
                 float* __restrict__ Y)
{
  __shared__ float pbuf[8][16][20];             // 20-float row pad: no bank conflicts
  const int lane = threadIdx.x & 31;
  const int wave = threadIdx.x >> 5;
  const int h  = blockIdx.x;
  const int l0 = blockIdx.y * 128 + wave * 16;
  const int half = lane >> 4;
  const int l16 = lane & 15;

  const float* Qh = Q  + (size_t)h * LSEQ * HDIM;
  const float* Kh = Kt + (size_t)h * HDIM * LSEQ;
  const float* Vh = V  + (size_t)h * LSEQ * HDIM;

  // Q fragments: 16 k-steps covering HDIM=64 (live across the whole loop)
  v2f qf[16];
#pragma unroll
  for (int s = 0; s < 16; ++s)
    qf[s] = *(const v2f*)(Qh + (size_t)(l0 + l16) * HDIM + s * 4 + 2 * half);

  v8f acc0{}, acc1{}, acc2{}, acc3{};
  float m_i[8], l_i[8];
#pragma unroll
  for (int r = 0; r < 8; ++r) { m_i[r] = -1e30f; l_i[r] = 0.0f; }

  for (int j0 = 0; j0 < LSEQ; j0 += 16) {
    // ---- hoist ALL K fragments: 32 loads in flight, one wait for 16 wmmas
    v2f kf[16];
#pragma unroll
    for (int kk = 0; kk < 16; ++kk) {
      const int t0 = kk * 4 + 2 * half;
      kf[kk].x = Kh[(size_t)t0 * LSEQ + j0 + l16];
      kf[kk].y = Kh[(size_t)(t0 + 1) * LSEQ + j0 + l16];
    }

    // ---- S = Q @ K  (16x16 tile) ----
    v8f s{};
#pragma unroll
    for (int kk = 0; kk < 16; ++kk)
      s = wmma4(qf[kk], kf[kk], s);

    // ---- issue V fragments NOW: latency hides behind softmax VALU below
    v2f vf[4][4];
#pragma unroll
    for (int ks = 0; ks < 4; ++ks) {
      const float* vb = Vh + (size_t)(j0 + ks * 4 + 2 * half) * HDIM;
#pragma unroll
      for (int t = 0; t < 4; ++t) {
        vf[ks][t].x = vb[t * 16 + l16];
        vf[ks][t].y = vb[HDIM + t * 16 + l16];
      }
    }

    // ---- + attention_mask ----
    const float* mrow = mask + (size_t)(l0 + 8 * half) * LSEQ + j0 + l16;
#pragma unroll
    for (int r = 0; r < 8; ++r) s[r] += mrow[(size_t)r * LSEQ];

    // ---- online softmax (row = one 16-lane half -> xor 1,2,4,8 reductions)
    float corr[8];
#pragma unroll
    for (int r = 0; r < 8; ++r) {
      float v = s[r];
#pragma unroll
      for (int off = 1; off < 16; off <<= 1)
        v = fmaxf(v, __shfl_xor(v, off, 32));        // row max
      float mn = fmaxf(m_i[r], v);
      float p  = __expf(s[r] - mn);
      float rs = p;
#pragma unroll
      for (int off = 1; off < 16; off <<= 1)
        rs += __shfl_xor(rs, off, 32);               // row sum
      corr[r] = __expf(m_i[r] - mn);
      l_i[r]  = l_i[r] * corr[r] + rs;
      m_i[r]  = mn;
      pbuf[wave][r + 8 * half][l16] = p;             // stash P for A-layout reload
    }

    // rescale accumulators by exp(m_old - m_new)
#pragma unroll
    for (int r = 0; r < 8; ++r) {
      acc0[r] *= corr[r]; acc1[r] *= corr[r];
      acc2[r] *= corr[r]; acc3[r] *= corr[r];
    }

    // ---- acc += P(16x16) @ V_block(16x64) ----
    // (wave-private LDS tile; same-wave DS ops are in-order -> no barrier)
#pragma unroll
    for (int ks = 0; ks < 4; ++ks) {
      v2f a = *(const v2f*)(&pbuf[wave][l16][ks * 4 + 2 * half]);
      acc0 = wmma4(a, vf[ks][0], acc0);
      acc1 = wmma4(a, vf[ks][1], acc1);
      acc2 = wmma4(a, vf[ks][2], acc2);
      acc3 = wmma4(a, vf[ks][3], acc3);
    }
  }

  // ---- normalize and scatter into (L, DMODEL) at column h*64 ----
  float inv[8];
#pragma unroll
  for (int r = 0; r < 8; ++r) inv[r] = 1.0f / l_i[r];
  float* yrow = Y + (size_t)(l0 + 8 * half) * DMODEL + h * HDIM + l16;
#pragma unroll
  for (int r = 0; r < 8; ++r) {
    yrow[0]  = acc0[r] * inv[r];
    yrow[16] = acc1[r] * inv[r];
    yrow[32] = acc2[r] * inv[r];
    yrow[48] = acc3[r] * inv[r];
    yrow += DMODEL;
  }
}

// ---------------------------------------------------------------------------
// Host-side orchestration (graph-capture safe: launches only).
// ---------------------------------------------------------------------------
extern "C" void kernel_launch(void* const* d_in, const int* in_sizes, int n_in,
                              void* d_out, int out_size, void* d_ws, size_t ws_size,
                              hipStream_t stream) {
  (void)in_sizes; (void)n_in; (void)out_size; (void)ws_size;

  const float* hs    = (const float*)d_in[0];   // (1, L, DIM)
  const float* cosb  = (const float*)d_in[1];   // (1, L, HD)
  const float* cosT  = (const float*)d_in[2];   // (1, HD, L)
  const float* sinb  = (const float*)d_in[3];   // (1, L, HD)
  const float* sinT  = (const float*)d_in[4];   // (1, HD, L)
  const float* amask = (const float*)d_in[5];   // (1, L, L)
  const float* kc    = (const float*)d_in[6];   // (H, HD, L)
  const float* vc    = (const float*)d_in[7];   // (H, L, HD)
  const float* kmask = (const float*)d_in[8];   // (1, 1, L)
  const float* vmask = (const float*)d_in[9];   // (1, L, 1)
  const float* wq    = (const float*)d_in[10];  // (DIM, DIM)
  const float* wk    = (const float*)d_in[11];
  const float* wv    = (const float*)d_in[12];
  const float* wo    = (const float*)d_in[13];

  const size_t LD = (size_t)LSEQ * DMODEL;      // 4,194,304 elements
  float* y_out = (float*)d_out;                 // (L, DIM)
  float* k_out = y_out + LD;                    // (H, HD, L)
  float* v_out = k_out + LD;                    // (H, L, HD)

  float* ws     = (float*)d_ws;                 // needs 4*LD floats = 64 MB
  float* q_raw  = ws;
  float* k_raw  = q_raw + LD;
  float* v_raw  = k_raw + LD;
  float* q_rope = v_raw + LD;
  float* y_mid  = k_raw;                        // reuse: k_raw dead after k-merge

  dim3 gemm_grid(LSEQ / 64, DMODEL / 256);      // (32, 8)
  dim3 elem_grid((NHEADS * LSEQ * HDIM) / 256); // 16384
  dim3 attn_grid(NHEADS, LSEQ / 128);           // (32, 16)

  // QKV projections: x @ W.T
  gemm_nt_f32<<<gemm_grid, 256, 0, stream>>>(hs, wq, q_raw, LSEQ, DMODEL, DMODEL);
  gemm_nt_f32<<<gemm_grid, 256, 0, stream>>>(hs, wk, k_raw, LSEQ, DMODEL, DMODEL);
  gemm_nt_f32<<<gemm_grid, 256, 0, stream>>>(hs, wv, v_raw, LSEQ, DMODEL, DMODEL);

  // RoPE + cache blends (k_out / v_out are final outputs 2 and 3)
  rope_q_kernel<<<elem_grid, 256, 0, stream>>>(q_raw, cosb, sinb, q_rope);
  rope_k_merge_kernel<<<elem_grid, 256, 0, stream>>>(k_raw, cosT, sinT, kc, kmask, k_out);
  v_merge_kernel<<<elem_grid, 256, 0, stream>>>(v_raw, vc, vmask, v_out);

  // Fused softmax attention -> y_mid (L, DIM)
  attn_kernel<<<attn_grid, 256, 0, stream>>>(q_rope, k_out, v_out, amask, y_mid);

  // Output projection: y_mid @ wo.T -> y_out
  gemm_nt_f32<<<gemm_grid, 256, 0, stream>>>(y_mid, wo, y_out, LSEQ, DMODEL, DMODEL);
}